// GCN_53764400611916
// MI455X (gfx1250) — compile-verified
//
#include <hip/hip_runtime.h>
#include <hip/hip_bf16.h>

#define N_NODES  100000
#define N_EDGES  3200000
#define N_GRAPHS 512
// F_IN=64, F_HID=16, F_OUT=2

typedef float v2f __attribute__((ext_vector_type(2)));
typedef float v8f __attribute__((ext_vector_type(8)));

// ---------------------------------------------------------------------------
// Zero a float region (harness poisons d_ws with 0xAA; we must re-zero every
// call since graph replays don't re-poison).
// ---------------------------------------------------------------------------
__global__ void gcn_zero_f32(float* __restrict__ p, long long n) {
    long long i = (long long)blockIdx.x * blockDim.x + threadIdx.x;
    if (i < n) p[i] = 0.0f;
}

// ---------------------------------------------------------------------------
// deg[col[e]] += ew[e]   (self-loop "+1" folded into dinv kernel)
// ---------------------------------------------------------------------------
__global__ void gcn_deg_edges(const long long* __restrict__ ei,
                              const float* __restrict__ ew,
                              float* __restrict__ deg) {
    int e = blockIdx.x * blockDim.x + threadIdx.x;
    if (e >= N_EDGES) return;
    int c = (int)ei[(long long)N_EDGES + e];
    atomicAdd(&deg[c], ew[e]);
}

__global__ void gcn_dinv(const float* __restrict__ deg, float* __restrict__ dinv) {
    int i = blockIdx.x * blockDim.x + threadIdx.x;
    if (i >= N_NODES) return;
    dinv[i] = rsqrtf(deg[i] + 1.0f);   // deg includes self-loop weight 1.0, always > 0
}

// ---------------------------------------------------------------------------
// H[N,16] = X[N,K] @ W[K,16] via V_WMMA_F32_16X16X4_F32.
// One wave per 16-row tile; K stepped by 4.
// A (16x4 f32): lane L<16 holds row M=L, K={kb*4+0, kb*4+1} in (a.x,a.y);
//               lane L>=16 holds row M=L-16, K={kb*4+2, kb*4+3}.  (ISA 7.12.2)
// B (4x16 f32): symmetric layout, lanes hold N, VGPRs hold K.
// C (16x16 f32): VGPR j -> M = j (+8 for lanes 16..31), N = lane&15.
// ---------------------------------------------------------------------------
template <int K>
__global__ void gcn_gemm_wmma(const float* __restrict__ X,
                              const float* __restrict__ W,
                              float* __restrict__ H) {
    const int wave = (int)((blockIdx.x * blockDim.x + threadIdx.x) >> 5);
    const int lane = threadIdx.x & 31;
    const int m0 = wave * 16;
    if (m0 >= N_NODES) return;            // whole-wave exit: EXEC stays all-ones for WMMA

    const int mrow  = m0 + (lane & 15);
    const int khalf = (lane & 16) ? 2 : 0;
    const int n     = lane & 15;

    v8f c = {};
#pragma unroll
    for (int kb = 0; kb < K / 4; ++kb) {
        const int k = kb * 4 + khalf;
        v2f a, b;
        a.x = X[mrow * K + k];
        a.y = X[mrow * K + k + 1];
        b.x = W[k * 16 + n];
        b.y = W[(k + 1) * 16 + n];
        // (neg_a, A, neg_b, B, c_mod, C, reuse_a, reuse_b)
        c = __builtin_amdgcn_wmma_f32_16x16x4_f32(false, a, false, b,
                                                  (short)0, c, false, false);
    }

    const int mbase = m0 + ((lane & 16) ? 8 : 0);
#pragma unroll
    for (int j = 0; j < 8; ++j)
        H[(mbase + j) * 16 + n] = c[j];
}

// ---------------------------------------------------------------------------
// Edge scatter: agg[col] += dinv[row]*ew*dinv[col] * h[row]
// One thread per (edge, float4 quad): coalesced b128 gather, 4 f32 atomics.
// ---------------------------------------------------------------------------
__global__ void gcn_scatter(const long long* __restrict__ ei,
                            const float* __restrict__ ew,
                            const float* __restrict__ dinv,
                            const float* __restrict__ h,
                            float* __restrict__ agg) {
    long long tid = (long long)blockIdx.x * blockDim.x + threadIdx.x;
    if (tid >= (long long)N_EDGES * 4) return;
    int e = (int)(tid >> 2);
    int q = (int)(tid & 3);
    int r = (int)ei[e];
    int c = (int)ei[(long long)N_EDGES + e];
    float nrm = dinv[r] * ew[e] * dinv[c];
    const float4 hv = ((const float4*)(h + (long long)r * 16))[q];
    float* ap = agg + (long long)c * 16 + q * 4;
    atomicAdd(ap + 0, nrm * hv.x);
    atomicAdd(ap + 1, nrm * hv.y);
    atomicAdd(ap + 2, nrm * hv.z);
    atomicAdd(ap + 3, nrm * hv.w);
}

// ---------------------------------------------------------------------------
// h[i,f] = relu(agg[i,f] + dinv[i]^2 * h[i,f] + b[f])   (self-loop fused)
// ---------------------------------------------------------------------------
__global__ void gcn_finish_conv(float* __restrict__ h,
                                const float* __restrict__ agg,
                                const float* __restrict__ dinv,
                                const float* __restrict__ b) {
    int tid = blockIdx.x * blockDim.x + threadIdx.x;
    if (tid >= N_NODES * 16) return;
    int i = tid >> 4, f = tid & 15;
    float di = dinv[i];
    float v = agg[tid] + di * di * h[tid] + b[f];
    h[tid] = v > 0.0f ? v : 0.0f;
}

// ---------------------------------------------------------------------------
// Global mean pool (sums + counts via atomics)
// ---------------------------------------------------------------------------
__global__ void gcn_pool(const float* __restrict__ h,
                         const long long* __restrict__ batch,
                         float* __restrict__ pooled,
                         float* __restrict__ counts) {
    int tid = blockIdx.x * blockDim.x + threadIdx.x;
    if (tid >= N_NODES * 16) return;
    int i = tid >> 4, f = tid & 15;
    int g = (int)batch[i];
    atomicAdd(&pooled[g * 16 + f], h[tid]);
    if (f == 0) atomicAdd(&counts[g], 1.0f);
}

// ---------------------------------------------------------------------------
// Head: mean, 16x2 GEMM, log_softmax over 2 classes
// ---------------------------------------------------------------------------
__global__ void gcn_head(const float* __restrict__ pooled,
                         const float* __restrict__ counts,
                         const float* __restrict__ Wfc,
                         const float* __restrict__ bfc,
                         float* __restrict__ out) {
    int g = blockIdx.x * blockDim.x + threadIdx.x;
    if (g >= N_GRAPHS) return;
    float inv = 1.0f / fmaxf(counts[g], 1.0f);
    float l0 = bfc[0], l1 = bfc[1];
#pragma unroll
    for (int k = 0; k < 16; ++k) {
        float m = pooled[g * 16 + k] * inv;
        l0 += m * Wfc[k * 2 + 0];
        l1 += m * Wfc[k * 2 + 1];
    }
    float mx  = fmaxf(l0, l1);
    float lse = mx + logf(expf(l0 - mx) + expf(l1 - mx));
    out[g * 2 + 0] = l0 - lse;
    out[g * 2 + 1] = l1 - lse;
}

extern "C" void kernel_launch(void* const* d_in, const int* in_sizes, int n_in,
                              void* d_out, int out_size, void* d_ws, size_t ws_size,
                              hipStream_t stream) {
    const float*     x     = (const float*)d_in[0];
    const long long* ei    = (const long long*)d_in[1];   // int64 [2, E]
    const float*     ew    = (const float*)d_in[2];
    const long long* batch = (const long long*)d_in[3];   // int64 [N]
    const float*     W1    = (const float*)d_in[4];
    const float*     b1    = (const float*)d_in[5];
    const float*     W2    = (const float*)d_in[6];
    const float*     b2    = (const float*)d_in[7];
    const float*     Wfc   = (const float*)d_in[8];
    const float*     bfc   = (const float*)d_in[9];
    float*           out   = (float*)d_out;

    // Workspace layout (floats). Zero-init region comes first, in one launch.
    float* ws     = (float*)d_ws;
    float* deg    = ws;                          // N
    float* agg1   = deg    + N_NODES;            // 16N
    float* agg2   = agg1   + 16 * N_NODES;       // 16N
    float* pooled = agg2   + 16 * N_NODES;       // 512*16
    float* counts = pooled + N_GRAPHS * 16;      // 512
    float* dinv   = counts + N_GRAPHS;           // N
    float* h1     = dinv   + N_NODES;            // 16N
    float* h2     = h1     + 16 * N_NODES;       // 16N

    const long long zcount = 33LL * N_NODES + N_GRAPHS * 17;  // deg+agg1+agg2+pooled+counts

    // 1) zero scratch that accumulates
    gcn_zero_f32<<<(int)((zcount + 255) / 256), 256, 0, stream>>>(ws, zcount);
    // 2) degree + normalization
    gcn_deg_edges<<<(N_EDGES + 255) / 256, 256, 0, stream>>>(ei, ew, deg);
    gcn_dinv<<<(N_NODES + 255) / 256, 256, 0, stream>>>(deg, dinv);
    // 3) layer 1: WMMA GEMM (K=64), scatter, fused self-loop+bias+relu
    {
        const int tiles = N_NODES / 16;               // 6250, exact
        const int blocks = (tiles + 7) / 8;           // 8 waves (256 thr) per block
        gcn_gemm_wmma<64><<<blocks, 256, 0, stream>>>(x, W1, h1);
    }
    gcn_scatter<<<(int)(((long long)N_EDGES * 4 + 255) / 256), 256, 0, stream>>>(ei, ew, dinv, h1, agg1);
    gcn_finish_conv<<<(N_NODES * 16 + 255) / 256, 256, 0, stream>>>(h1, agg1, dinv, b1);
    // 4) layer 2: WMMA GEMM (K=16), scatter, fused epilogue
    {
        const int tiles = N_NODES / 16;
        const int blocks = (tiles + 7) / 8;
        gcn_gemm_wmma<16><<<blocks, 256, 0, stream>>>(h1, W2, h2);
    }
    gcn_scatter<<<(int)(((long long)N_EDGES * 4 + 255) / 256), 256, 0, stream>>>(ei, ew, dinv, h2, agg2);
    gcn_finish_conv<<<(N_NODES * 16 + 255) / 256, 256, 0, stream>>>(h2, agg2, dinv, b2);
    // 5) pool + head
    gcn_pool<<<(N_NODES * 16 + 255) / 256, 256, 0, stream>>>(h2, batch, pooled, counts);
    gcn_head<<<(N_GRAPHS + 255) / 256, 256, 0, stream>>>(pooled, counts, Wfc, bfc, out);
}